// TopK_with_h_40200893890652
// MI455X (gfx1250) — compile-verified
//
#include <hip/hip_runtime.h>
#include <hip/hip_bf16.h>

// ---------------------------------------------------------------------------
// Problem constants (from reference): B=64, N=4096, F=512, R=1024, K=64, G=512
// ---------------------------------------------------------------------------
#define BATCH   64
#define NNODES  4096
#define FDIM    512
#define RDIM    1024
#define TOPK    64
#define GDIM    512

typedef __attribute__((ext_vector_type(2))) float v2f;
typedef __attribute__((ext_vector_type(4))) float v4f;
typedef __attribute__((ext_vector_type(8))) float v8f;

// ---------------------------------------------------------------------------
// Kernel 1: scorer = tanh(h_selector @ W_map^T + b_map)   [64 x 512]
// True GEMM -> V_WMMA_F32_16X16X4_F32. One wave per 16x16 C tile.
// A = h [M=64 x K=1024] row-major; B[k][n] = W[n][k] (NT gemm), W [512 x 1024].
// A frag (16x4 f32, ISA layout): lanes 0-15: VGPR0=K0,VGPR1=K1; lanes 16-31: K2,K3.
// ---------------------------------------------------------------------------
__global__ __launch_bounds__(32) void scorer_wmma_kernel(
    const float* __restrict__ H,      // [64,1024]
    const float* __restrict__ W,      // [512,1024]
    const float* __restrict__ bias,   // [512]
    float* __restrict__ scorer_ws,    // [64,512] workspace copy
    float* __restrict__ scorer_out)   // [64,512] output #3
{
    const int tile = blockIdx.x;           // 0..127 = (4 M-tiles) x (32 N-tiles)
    const int tm   = (tile & 3)  * 16;     // batch tile base
    const int tn   = (tile >> 2) * 16;     // feature tile base
    const int lane = threadIdx.x;          // 0..31 (wave32)
    const int half = lane >> 4;            // 0: K pair {0,1}; 1: K pair {2,3}
    const int l16  = lane & 15;

    const float* Arow = H + (size_t)(tm + l16) * RDIM;  // this lane's m-row
    const float* Brow = W + (size_t)(tn + l16) * RDIM;  // this lane's n-row (NT)

    v8f c = {};
    for (int k0 = 0; k0 < RDIM; k0 += 4) {
        const int kk = k0 + half * 2;
        v2f a, b;
        a.x = Arow[kk]; a.y = Arow[kk + 1];
        b.x = Brow[kk]; b.y = Brow[kk + 1];
        // D = A(16x4) * B(4x16) + C ; 8 args: neg_a, A, neg_b, B, c_mod, C, reuse_a, reuse_b
        c = __builtin_amdgcn_wmma_f32_16x16x4_f32(false, a, false, b,
                                                  (short)0, c, false, false);
    }

    // D layout: VGPR v -> row m=v (lanes 0-15) / m=v+8 (lanes 16-31), col = lane&15
    const int   col = l16;
    const float bn  = bias[tn + col];
    #pragma unroll
    for (int v = 0; v < 8; ++v) {
        const int gm  = tm + v + half * 8;
        const float s = tanhf(c[v] + bn);
        scorer_ws [(size_t)gm * FDIM + tn + col] = s;
        scorer_out[(size_t)gm * FDIM + tn + col] = s;
    }
}

// ---------------------------------------------------------------------------
// Kernel 2: inv_norm[b] = 1/||scorer[b]||  (deterministic block reduction)
// ---------------------------------------------------------------------------
__global__ __launch_bounds__(256) void norm_kernel(
    const float* __restrict__ scorer, float* __restrict__ inv_norm)
{
    __shared__ float rv[256];
    const int b = blockIdx.x, t = threadIdx.x;
    const float a0 = scorer[(size_t)b * FDIM + t];
    const float a1 = scorer[(size_t)b * FDIM + 256 + t];
    rv[t] = a0 * a0 + a1 * a1;
    __syncthreads();
    for (int off = 128; off > 0; off >>= 1) {
        if (t < off) rv[t] += rv[t + off];
        __syncthreads();
    }
    if (t == 0) inv_norm[b] = rsqrtf(rv[0]);
}

// ---------------------------------------------------------------------------
// Kernel 3: scores[b,n] = dot(node_embs[b,n,:], scorer[b,:]) * inv_norm[b]
// THE bandwidth kernel: one streaming NT pass over 536 MB (> 192 MB L2, so
// TH=NT avoids thrashing). Each wave keeps its scorer fragment in registers
// (16 VGPRs) and processes 8 consecutive rows -> scorer re-load traffic /8.
// ---------------------------------------------------------------------------
#define ROWS_PER_WAVE 8
__global__ __launch_bounds__(256) void scores_kernel(
    const float* __restrict__ embs,     // [64,4096,512]
    const float* __restrict__ scorer,   // [64,512]
    const float* __restrict__ inv_norm, // [64]
    float* __restrict__ scores)         // [64,4096]
{
    const int wave = (int)((blockIdx.x * blockDim.x + threadIdx.x) >> 5);
    const int lane = threadIdx.x & 31;
    const int row0 = wave * ROWS_PER_WAVE;        // 8 rows, all in same batch
    const int b    = row0 >> 12;                  // /4096

    // scorer fragment for this wave: 16 floats/lane, covers all 512 features
    const v4f* srow = reinterpret_cast<const v4f*>(scorer + (size_t)b * FDIM);
    v4f s0 = srow[lane +  0], s1 = srow[lane + 32],
        s2 = srow[lane + 64], s3 = srow[lane + 96];
    const float inv = inv_norm[b];

    #pragma unroll 2
    for (int r = 0; r < ROWS_PER_WAVE; ++r) {
        const int row = row0 + r;
        const v4f* erow = reinterpret_cast<const v4f*>(embs + (size_t)row * FDIM);
        const v4f e0 = __builtin_nontemporal_load(&erow[lane +  0]);
        const v4f e1 = __builtin_nontemporal_load(&erow[lane + 32]);
        const v4f e2 = __builtin_nontemporal_load(&erow[lane + 64]);
        const v4f e3 = __builtin_nontemporal_load(&erow[lane + 96]);
        const v4f p  = e0 * s0 + e1 * s1 + e2 * s2 + e3 * s3;
        float acc = p.x + p.y + p.z + p.w;
        #pragma unroll
        for (int off = 16; off > 0; off >>= 1)
            acc += __shfl_xor(acc, off, 32);
        if (lane == 0) scores[row] = acc * inv;
    }
}

// ---------------------------------------------------------------------------
// Kernel 4: per batch row -- softmax stats (lse, entropy), iterative top-64,
// policy log-prob, tiled index output. Block per batch, scores row in LDS.
// ---------------------------------------------------------------------------
__global__ __launch_bounds__(256) void topk_kernel(
    const float* __restrict__ scores,  // [64,4096]
    float* __restrict__ out_policy,    // [64]
    float* __restrict__ out_entropy,   // [64]
    float* __restrict__ out_idx,       // [64,512] (written as float)
    int*   __restrict__ ws_idx,        // [64,64]
    float* __restrict__ ws_scale)      // [64,64]  tanh(score at topk)
{
    __shared__ float s[NNODES];
    __shared__ float rv[256];
    __shared__ int   ri[256];
    __shared__ float sel_val[TOPK];
    __shared__ int   sel_idx[TOPK];

    const int b = blockIdx.x, t = threadIdx.x;
    const float* row = scores + (size_t)b * NNODES;
    for (int i = t; i < NNODES; i += 256) s[i] = row[i];
    __syncthreads();

    // max
    float mx = -__builtin_inff();
    for (int i = t; i < NNODES; i += 256) mx = fmaxf(mx, s[i]);
    rv[t] = mx; __syncthreads();
    for (int off = 128; off > 0; off >>= 1) {
        if (t < off) rv[t] = fmaxf(rv[t], rv[t + off]);
        __syncthreads();
    }
    mx = rv[0]; __syncthreads();

    // S1 = sum exp(s-mx), S2 = sum exp(s-mx)*s
    float s1 = 0.f, s2 = 0.f;
    for (int i = t; i < NNODES; i += 256) {
        const float e = __expf(s[i] - mx);
        s1 += e; s2 += e * s[i];
    }
    rv[t] = s1; __syncthreads();
    for (int off = 128; off > 0; off >>= 1) { if (t < off) rv[t] += rv[t + off]; __syncthreads(); }
    s1 = rv[0]; __syncthreads();
    rv[t] = s2; __syncthreads();
    for (int off = 128; off > 0; off >>= 1) { if (t < off) rv[t] += rv[t + off]; __syncthreads(); }
    s2 = rv[0]; __syncthreads();
    const float lse = mx + __logf(s1);

    // iterative argmax selection x64 (smaller-index tie break)
    for (int k = 0; k < TOPK; ++k) {
        float bv = -__builtin_inff(); int bi = NNODES;
        for (int i = t; i < NNODES; i += 256) {
            const float v = s[i];
            if (v > bv) { bv = v; bi = i; }
        }
        rv[t] = bv; ri[t] = bi; __syncthreads();
        for (int off = 128; off > 0; off >>= 1) {
            if (t < off) {
                const float ov = rv[t + off]; const int oi = ri[t + off];
                if (ov > rv[t] || (ov == rv[t] && oi < ri[t])) { rv[t] = ov; ri[t] = oi; }
            }
            __syncthreads();
        }
        if (t == 0) { sel_val[k] = rv[0]; sel_idx[k] = ri[0]; s[ri[0]] = -__builtin_inff(); }
        __syncthreads();
    }

    if (t < TOPK) {
        const float v = sel_val[t]; const int id = sel_idx[t];
        ws_idx  [b * TOPK + t] = id;
        ws_scale[b * TOPK + t] = tanhf(v);
        #pragma unroll
        for (int r = 0; r < GDIM / TOPK; ++r)       // idx tiled x8 to G=512
            out_idx[(size_t)b * GDIM + r * TOPK + t] = (float)id;
    }
    if (t == 0) {
        float pol = 0.f;
        for (int k = 0; k < TOPK; ++k) pol += sel_val[k] - lse;
        out_policy [b] = pol / (float)TOPK;         // mean over 512 tiled == mean over 64
        out_entropy[b] = lse - s2 / s1;
    }
}

// ---------------------------------------------------------------------------
// Kernel 5: gather + scale + transpose. out[b,f,g] = embs[b, idx[g%64], f] *
// tanh_scale[g%64]. Block = (batch, 64-wide f-chunk). 64x64 LDS tile with
// pad-65 stride -> conflict-free reads; NT float2 stores (64 MB, never
// re-read) coalesced over g.
// ---------------------------------------------------------------------------
__global__ __launch_bounds__(256) void gather_kernel(
    const float* __restrict__ embs,     // [64,4096,512]
    const int*   __restrict__ ws_idx,   // [64,64]
    const float* __restrict__ ws_scale, // [64,64]
    float* __restrict__ out)            // [64,512,512]
{
    __shared__ float tile[TOPK][65];
    __shared__ int   ridx[TOPK];
    __shared__ float rscale[TOPK];

    const int blk = blockIdx.x;
    const int b   = blk >> 3;            // 8 f-chunks of 64
    const int f0  = (blk & 7) * 64;
    const int t   = threadIdx.x;

    if (t < TOPK) { ridx[t] = ws_idx[b * TOPK + t]; rscale[t] = ws_scale[b * TOPK + t]; }
    __syncthreads();

    // load 64 rows x 64 features into LDS
    for (int e = t; e < TOPK * 64; e += 256) {
        const int r = e >> 6, f = e & 63;
        tile[r][f] = embs[((size_t)b * NNODES + ridx[r]) * FDIM + f0 + f];
    }
    __syncthreads();

    // write: for each f in chunk, all 512 g contiguous (float2 per thread)
    const int g0 = t * 2;
    const float sx = rscale[g0 & 63], sy = rscale[(g0 + 1) & 63];
    for (int f = 0; f < 64; ++f) {
        v2f v;
        v.x = tile[ g0      & 63][f] * sx;
        v.y = tile[(g0 + 1) & 63][f] * sy;
        __builtin_nontemporal_store(v,
            reinterpret_cast<v2f*>(&out[((size_t)b * FDIM + f0 + f) * GDIM + g0]));
    }
}

// ---------------------------------------------------------------------------
// Launch: outputs concatenated flat in reference return order:
//   topk_embs^T [64*512*512] | score_policy [64] | scorer [64*512] |
//   entropy [64] | idx [64*512]
// ---------------------------------------------------------------------------
extern "C" void kernel_launch(void* const* d_in, const int* in_sizes, int n_in,
                              void* d_out, int out_size, void* d_ws, size_t ws_size,
                              hipStream_t stream) {
    const float* node_embs = (const float*)d_in[0];
    // d_in[1] = mask (unused, as in reference)
    const float* h_sel     = (const float*)d_in[2];
    const float* W_map     = (const float*)d_in[3];
    const float* b_map     = (const float*)d_in[4];

    float* out          = (float*)d_out;
    float* out_topk     = out;                                   // 16,777,216
    float* out_policy   = out + (size_t)BATCH * FDIM * GDIM;     // 64
    float* out_scorer   = out_policy + BATCH;                    // 32,768
    float* out_entropy  = out_scorer + (size_t)BATCH * FDIM;     // 64
    float* out_idx      = out_entropy + BATCH;                   // 32,768

    float* ws           = (float*)d_ws;
    float* ws_scorer    = ws;                                    // 32,768
    float* ws_invnorm   = ws_scorer + (size_t)BATCH * FDIM;      // 64
    float* ws_scores    = ws_invnorm + BATCH;                    // 262,144
    int*   ws_idx       = (int*)(ws_scores + (size_t)BATCH * NNODES); // 4,096
    float* ws_scale     = (float*)ws_idx + (size_t)BATCH * TOPK; // 4,096

    // 1) scorer GEMM via f32 WMMA (128 tiles, 1 wave each)
    scorer_wmma_kernel<<<(BATCH / 16) * (FDIM / 16), 32, 0, stream>>>(
        h_sel, W_map, b_map, ws_scorer, out_scorer);

    // 2) per-batch L2 norm (deterministic reduction)
    norm_kernel<<<BATCH, 256, 0, stream>>>(ws_scorer, ws_invnorm);

    // 3) streaming scores pass (the 536 MB NT read): wave per 8 rows
    scores_kernel<<<(BATCH * NNODES) / (8 * ROWS_PER_WAVE), 256, 0, stream>>>(
        node_embs, ws_scorer, ws_invnorm, ws_scores);

    // 4) top-64 + softmax stats per batch
    topk_kernel<<<BATCH, 256, 0, stream>>>(
        ws_scores, out_policy, out_entropy, out_idx, ws_idx, ws_scale);

    // 5) gather + scale + transpose into [B, F, G]
    gather_kernel<<<BATCH * (FDIM / 64), 256, 0, stream>>>(
        node_embs, ws_idx, ws_scale, out_topk);
}